// Attn_time_47863115547247
// MI455X (gfx1250) — compile-verified
//
#include <hip/hip_runtime.h>

#define T_DIM   1024      // time table is T_DIM x T_DIM
#define SEQ_LEN 8192      // his length (row length of output)
#define BLOCK   256       // 8 wave32s
#define CH      8         // float4 chunks per thread: 8192 / (256*4)
#define NWAVES  (BLOCK / 32)

typedef __attribute__((ext_vector_type(4))) unsigned int u32x4;
typedef __attribute__((ext_vector_type(8))) int          i32x8;
typedef __attribute__((ext_vector_type(4))) int          i32x4;
typedef __attribute__((ext_vector_type(4))) float        f32x4;

__device__ __forceinline__ float wave_max(float x) {
#pragma unroll
  for (int o = 16; o > 0; o >>= 1) x = fmaxf(x, __shfl_xor(x, o, 32));
  return x;
}
__device__ __forceinline__ float wave_sum(float x) {
#pragma unroll
  for (int o = 16; o > 0; o >>= 1) x += __shfl_xor(x, o, 32);
  return x;
}

__global__ __launch_bounds__(BLOCK) void attn_time_softmax(
    const int* __restrict__ his, const int* __restrict__ cur,
    const float* __restrict__ mat, float* __restrict__ out)
{
  __shared__ float row[T_DIM];       // 4 KB staged table row
  __shared__ float redm[NWAVES];
  __shared__ float reds[NWAVES];

  const int tid = threadIdx.x;
  const int r   = blockIdx.x;
  const int c   = cur[r];                       // wave-uniform
  const float* src = mat + (size_t)c * T_DIM;

#if defined(__gfx1250__) && __has_builtin(__builtin_amdgcn_tensor_load_to_lds)
  // --- TDM: DMA the 4KB table row into LDS (wave 0 only; TDM ignores EXEC) ---
  if (tid < 32) {
    const unsigned           lds_off = (unsigned)(size_t)(void*)row; // low 32 bits of generic ptr = LDS byte offset
    const unsigned long long ga      = (unsigned long long)(size_t)src;

    u32x4 g0;
    g0.x = 1u;                                              // count=1, user mode, no gather
    g0.y = lds_off;                                         // lds_addr [63:32]
    g0.z = (unsigned)ga;                                    // global_addr [95:64]
    g0.w = ((unsigned)(ga >> 32) & 0x01FFFFFFu)             // global_addr [120:96]
         | 0x80000000u;                                     // type=2 ("image") [127:126]

    i32x8 g1;
    g1[0] = (int)(2u << 16);                 // workgroup_mask=0, data_size=2 (4B), no barrier/iter/pad
    g1[1] = (int)((unsigned)T_DIM << 16);    // tensor_dim0 = 1024 (low 16)
    g1[2] = (int)(1u << 16);                 // tensor_dim0 hi=0 | tensor_dim1 = 1
    g1[3] = (int)((unsigned)T_DIM << 16);    // tensor_dim1 hi=0 | tile_dim0 = 1024
    g1[4] = 1;                               // tile_dim1 = 1, tile_dim2 = 0
    g1[5] = T_DIM;                           // tensor_dim0_stride = 1024 (low 32)
    g1[6] = (int)((unsigned)T_DIM << 16);    // stride0 hi=0 | tensor_dim1_stride low16 = 1024
    g1[7] = 0;

    i32x4 g2 = {0, 0, 0, 0};
    i32x4 g3 = {0, 0, 0, 0};
#if __clang_major__ >= 23
    i32x8 g4 = {0, 0, 0, 0, 0, 0, 0, 0};
    __builtin_amdgcn_tensor_load_to_lds(g0, g1, g2, g3, g4, 0);
#else
    __builtin_amdgcn_tensor_load_to_lds(g0, g1, g2, g3, 0);
#endif
  }
#else
  // Fallback: cooperative vector copy (256 threads x 16B = 4KB)
  ((f32x4*)row)[tid] = ((const f32x4*)src)[tid];
#endif

  // --- Overlap with DMA: fetch this thread's 32 gather indices (8 x b128, coalesced) ---
  const i32x4* his4 = (const i32x4*)his;
  i32x4 h[CH];
#pragma unroll
  for (int k = 0; k < CH; ++k) h[k] = his4[k * BLOCK + tid];

#if defined(__gfx1250__) && __has_builtin(__builtin_amdgcn_tensor_load_to_lds)
  if (tid < 32) __builtin_amdgcn_s_wait_tensorcnt(0);
#endif
  __syncthreads();

  // --- Gather from LDS + running max ---
  f32x4 v[CH];
  float m = -__builtin_inff();
#pragma unroll
  for (int k = 0; k < CH; ++k) {
    f32x4 e;
    e.x = row[h[k].x];
    e.y = row[h[k].y];
    e.z = row[h[k].z];
    e.w = row[h[k].w];
    v[k] = e;
    m = fmaxf(m, fmaxf(fmaxf(e.x, e.y), fmaxf(e.z, e.w)));
  }

  // --- Block-wide max (wave32 shuffle + 8-slot LDS) ---
  m = wave_max(m);
  if ((tid & 31) == 0) redm[tid >> 5] = m;
  __syncthreads();
  float bm = redm[0];
#pragma unroll
  for (int w = 1; w < NWAVES; ++w) bm = fmaxf(bm, redm[w]);

  // --- exp + block-wide sum ---
  float s = 0.f;
#pragma unroll
  for (int k = 0; k < CH; ++k) {
    f32x4 e = v[k];
    e.x = __expf(e.x - bm);
    e.y = __expf(e.y - bm);
    e.z = __expf(e.z - bm);
    e.w = __expf(e.w - bm);
    v[k] = e;
    s += (e.x + e.y) + (e.z + e.w);
  }
  s = wave_sum(s);
  if ((tid & 31) == 0) reds[tid >> 5] = s;
  __syncthreads();
  float bs = reds[0];
#pragma unroll
  for (int w = 1; w < NWAVES; ++w) bs += reds[w];
  const float inv = 1.0f / bs;

  // --- Streaming non-temporal b128 stores (don't evict the hot 4MB table) ---
  f32x4* out4 = (f32x4*)(out + (size_t)r * SEQ_LEN);
#pragma unroll
  for (int k = 0; k < CH; ++k) {
    f32x4 e = v[k];
    e.x *= inv; e.y *= inv; e.z *= inv; e.w *= inv;
    __builtin_nontemporal_store(e, out4 + k * BLOCK + tid);
  }
}

extern "C" void kernel_launch(void* const* d_in, const int* in_sizes, int n_in,
                              void* d_out, int out_size, void* d_ws, size_t ws_size,
                              hipStream_t stream) {
  const int*   his = (const int*)d_in[0];    // (8192,) int32
  const int*   cur = (const int*)d_in[1];    // (4096,) int32
  const float* mat = (const float*)d_in[2];  // (1024,1024) f32
  float*       out = (float*)d_out;          // (4096,8192) f32

  const int rows = in_sizes[1];              // STATE_LEN = 4096
  attn_time_softmax<<<dim3(rows), dim3(BLOCK), 0, stream>>>(his, cur, mat, out);
}